// ious_3d_loss_56453050139232
// MI455X (gfx1250) — compile-verified
//
#include <hip/hip_runtime.h>
#include <math.h>

// 3D IoU loss (mean(1 - iou3d)) for N paired rotated boxes [x,y,z,h,w,l,ry].
// Fully register-resident per-pair geometry:
//  - 24 candidate points (16 edge-edge intersections + 8 corners) in VGPRs
//  - convex-hull ordering via cyclic-successor scan (static indices only, no
//    scratch, no sort) using a diamond pseudo-angle (order-equivalent to atan2;
//    the cyclic shoelace sum is invariant to the branch-cut rotation)
//  - deterministic two-stage LDS reduction (bitwise reproducible, no atomics)
// No GEMM structure exists here -> WMMA/TDM not applicable; optimize wave32 VALU.

#define EPS_F 1e-8f
#define BIGA  3.0e38f

__global__ __launch_bounds__(256)
void iou3d_loss_main(const float* __restrict__ g,
                     const float* __restrict__ q,
                     float* __restrict__ partial,
                     int n) {
    const int i = blockIdx.x * blockDim.x + threadIdx.x;
    float loss = 0.0f;

    if (i < n) {
        const float* A = g + 7 * i;
        const float* B = q + 7 * i;
        const float ax = A[0], ay = A[1], az = A[2], ah = A[3], aw = A[4], al = A[5], ar = A[6];
        const float bx = B[0], by = B[1], bz = B[2], bh = B[3], bw = B[4], bl = B[5], br = B[6];

        float sa, ca, sb, cb;
        sincosf(ar, &sa, &ca);
        sincosf(br, &sb, &cb);

        // BEV corners: lx=[.5,.5,-.5,-.5]*l, lz=[.5,-.5,-.5,.5]*w
        // X = c*cx + s*cz + x ; Z = -s*cx + c*cz + z
        const float LX[4] = { 0.5f, 0.5f, -0.5f, -0.5f };
        const float LZ[4] = { 0.5f, -0.5f, -0.5f, 0.5f };
        float cax[4], caz[4], cbx[4], cbz[4];
        #pragma unroll
        for (int k = 0; k < 4; ++k) {
            float cx = LX[k] * al, cz = LZ[k] * aw;
            cax[k] =  ca * cx + sa * cz + ax;
            caz[k] = -sa * cx + ca * cz + az;
            cx = LX[k] * bl; cz = LZ[k] * bw;
            cbx[k] =  cb * cx + sb * cz + bx;
            cbz[k] = -sb * cx + cb * cz + bz;
        }

        // Candidate points: 16 edge-edge intersections, 4 A-corners, 4 B-corners.
        float px[24], pz[24];
        bool  vb[24];

        #pragma unroll
        for (int e = 0; e < 4; ++e) {
            const float a1x = cax[e], a1z = caz[e];
            const float d1x = cax[(e + 1) & 3] - a1x;
            const float d1z = caz[(e + 1) & 3] - a1z;
            #pragma unroll
            for (int f = 0; f < 4; ++f) {
                const float b1x = cbx[f], b1z = cbz[f];
                const float d2x = cbx[(f + 1) & 3] - b1x;
                const float d2z = cbz[(f + 1) & 3] - b1z;
                const float denom = d1x * d2z - d1z * d2x;
                const bool  nz    = fabsf(denom) > EPS_F;
                const float safe  = nz ? denom : 1.0f;
                const float rx = b1x - a1x, rz = b1z - a1z;
                const float t = (rx * d2z - rz * d2x) / safe;
                const float u = (rx * d1z - rz * d1x) / safe;
                const int idx = e * 4 + f;
                px[idx] = a1x + t * d1x;
                pz[idx] = a1z + t * d1z;
                vb[idx] = nz && t >= 0.0f && t <= 1.0f && u >= 0.0f && u <= 1.0f;
            }
        }

        #pragma unroll
        for (int k = 0; k < 4; ++k) {
            // A-corner inside B?
            px[16 + k] = cax[k]; pz[16 + k] = caz[k];
            float rx = cax[k] - bx, rz = caz[k] - bz;
            float lx = cb * rx - sb * rz;
            float lz = sb * rx + cb * rz;
            vb[16 + k] = (fabsf(lx) <= 0.5f * bl + 1e-5f) && (fabsf(lz) <= 0.5f * bw + 1e-5f);
            // B-corner inside A?
            px[20 + k] = cbx[k]; pz[20 + k] = cbz[k];
            rx = cbx[k] - ax; rz = cbz[k] - az;
            lx = ca * rx - sa * rz;
            lz = sa * rx + ca * rz;
            vb[20 + k] = (fabsf(lx) <= 0.5f * al + 1e-5f) && (fabsf(lz) <= 0.5f * aw + 1e-5f);
        }

        // Mean of masked points (cnt clamped to >= 1).
        float sx = 0.0f, sz = 0.0f; int cnt = 0;
        #pragma unroll
        for (int k = 0; k < 24; ++k) {
            sx += vb[k] ? px[k] : 0.0f;
            sz += vb[k] ? pz[k] : 0.0f;
            cnt += vb[k] ? 1 : 0;
        }
        const float inv_c = 1.0f / (float)(cnt > 0 ? cnt : 1);
        const float mx = sx * inv_c, mz = sz * inv_c;

        // Diamond pseudo-angle: monotone in atan2 over (-pi,pi]; order-preserving
        // for distinct directions. (0,0) -> 0 to match atan2(0,0)=0.
        float ang[24];
        #pragma unroll
        for (int k = 0; k < 24; ++k) {
            const float rx = px[k] - mx;
            const float rz = pz[k] - mz;
            const float den = fabsf(rx) + fabsf(rz);
            const float p   = (den > 0.0f) ? (rx / den) : 1.0f;   // p=1 -> d=0
            ang[k] = (rz < 0.0f) ? (p - 1.0f) : (1.0f - p);       // in [-2, 2]
        }

        // Wrap point: valid point with minimal (angle, index) key.
        float wA = BIGA, wX = 0.0f, wZ = 0.0f;
        #pragma unroll
        for (int j = 0; j < 24; ++j) {
            const bool upd = vb[j] && (ang[j] < wA);  // strict < keeps earliest index
            wA = upd ? ang[j] : wA;
            wX = upd ? px[j]  : wX;
            wZ = upd ? pz[j]  : wZ;
        }

        // Cyclic-successor shoelace: for each valid point, cross with the valid
        // point of next-larger (angle, index) key; wrap to the global minimum.
        // Equals the reference's sort->replace-invalid->roll shoelace exactly
        // (invalid points contribute 0; cyclic sum is rotation-invariant).
        float acc = 0.0f;
        #pragma unroll
        for (int c0 = 0; c0 < 24; ++c0) {
            float bA = BIGA, bX = 0.0f, bZ = 0.0f;
            #pragma unroll
            for (int j = 0; j < 24; ++j) {
                if (j == c0) continue;
                // lexicographic (angle, index) "after c0": index compare is
                // compile-time, so it collapses to >= vs > on the angle.
                const bool after = (j > c0) ? (ang[j] >= ang[c0]) : (ang[j] > ang[c0]);
                const bool upd   = vb[j] && after && (ang[j] < bA);
                bA = upd ? ang[j] : bA;
                bX = upd ? px[j]  : bX;
                bZ = upd ? pz[j]  : bZ;
            }
            const bool  has = bA < 3.0e37f;
            const float nxp = has ? bX : wX;
            const float nzp = has ? bZ : wZ;
            const float cr  = px[c0] * nzp - pz[c0] * nxp;
            acc += vb[c0] ? cr : 0.0f;
        }
        const float area = 0.5f * fabsf(acc);

        // Height overlap + 3D IoU.
        const float h_ov  = fmaxf(fminf(ay, by) - fmaxf(ay - ah, by - bh), 0.0f);
        const float inter = area * h_ov;
        const float vol_a = ah * aw * al;
        const float vol_b = bh * bw * bl;
        float iou = inter / (vol_a + vol_b - inter + EPS_F);
        iou = fminf(fmaxf(iou, 0.0f), 1.0f);
        loss = 1.0f - iou;
    }

    // Deterministic block reduction (LDS tree), one partial per block.
    __shared__ float sm[256];
    sm[threadIdx.x] = loss;
    __syncthreads();
    #pragma unroll
    for (int s = 128; s > 0; s >>= 1) {
        if ((int)threadIdx.x < s) sm[threadIdx.x] += sm[threadIdx.x + s];
        __syncthreads();
    }
    if (threadIdx.x == 0) partial[blockIdx.x] = sm[0];
}

__global__ __launch_bounds__(256)
void iou3d_loss_reduce(const float* __restrict__ partial, int nblocks,
                       float* __restrict__ out, float invN) {
    __shared__ float sm[256];
    float s = 0.0f;
    for (int i = threadIdx.x; i < nblocks; i += 256) s += partial[i];
    sm[threadIdx.x] = s;
    __syncthreads();
    #pragma unroll
    for (int st = 128; st > 0; st >>= 1) {
        if ((int)threadIdx.x < st) sm[threadIdx.x] += sm[threadIdx.x + st];
        __syncthreads();
    }
    if (threadIdx.x == 0) out[0] = sm[0] * invN;
}

extern "C" void kernel_launch(void* const* d_in, const int* in_sizes, int n_in,
                              void* d_out, int out_size, void* d_ws, size_t ws_size,
                              hipStream_t stream) {
    const float* gboxes = (const float*)d_in[0];
    const float* qboxes = (const float*)d_in[1];
    const int n = in_sizes[0] / 7;               // 524288

    const int block = 256;                       // 8 wave32 waves per block
    const int nblocks = (n + block - 1) / block; // 2048 partials (8 KB of d_ws)
    float* partial = (float*)d_ws;

    iou3d_loss_main<<<nblocks, block, 0, stream>>>(gboxes, qboxes, partial, n);
    iou3d_loss_reduce<<<1, 256, 0, stream>>>(partial, nblocks, (float*)d_out,
                                             1.0f / (float)n);
}